// GridToGridConv_62758062129886
// MI455X (gfx1250) — compile-verified
//
#include <hip/hip_runtime.h>
#include <math.h>

#define N_NODES 50000
#define N_EDGES 800000
#define FDIM 128
#define LSTR 132   // LDS stride for 16-row A tiles (floats)
#define WSTR 136   // LDS stride for 64-row W chunks (disjoint banks for K halves)

typedef float v2f __attribute__((ext_vector_type(2)));
typedef float v8f __attribute__((ext_vector_type(8)));

// ---------------------------------------------------------------- WMMA helper
#if __has_builtin(__builtin_amdgcn_wmma_f32_16x16x4_f32)
__device__ __forceinline__ v8f wmma4(v2f a, v2f b, v8f c) {
  // D = A(16x4 f32) * B(4x16 f32) + C(16x16 f32)
  return __builtin_amdgcn_wmma_f32_16x16x4_f32(false, a, false, b, (short)0, c,
                                               false, false);
}
#else
__device__ __forceinline__ v8f wmma4(v2f a, v2f b, v8f c) {
  c[0] += a.x * b.x + a.y * b.y;
  return c;
}
#endif

__device__ __forceinline__ float elu1(float v) {
  return v > 0.f ? v : (expf(v) - 1.f);
}

// stage 64 rows x 128 cols of W (global, ld=128) into LDS with stride WSTR
__device__ __forceinline__ void stage_w64(float* wl, const float* __restrict__ W,
                                          int krow0, int nrows, int t) {
  const int n4 = nrows * 32;  // float4 items
  for (int idx = t; idx < n4; idx += 256) {
    const int row = idx >> 5, c4 = idx & 31;
    ((float4*)&wl[row * WSTR])[c4] =
        ((const float4*)&W[(krow0 + row) * FDIM])[c4];
  }
}

// ------------------------------------------------ GEMM1: h1 = x @ W1 + b1
__global__ __launch_bounds__(256) void k_gemm_h1(const float* __restrict__ x,
                                                 const float* __restrict__ W1,
                                                 const float* __restrict__ b1,
                                                 float* __restrict__ h1) {
  __shared__ float xs[16 * LSTR];
  __shared__ float wl[64 * WSTR];
  const int t = threadIdx.x;
  const int row0 = blockIdx.x * 16;
#pragma unroll
  for (int i = 0; i < 8; ++i) {
    int idx = t + i * 256;
    int r = idx >> 7, c = idx & 127;
    xs[r * LSTR + c] = x[(row0 + r) * FDIM + c];
  }
  const int wave = t >> 5;
  const int lane = t & 31;
  const int n = lane & 15;     // A row (M) and B col (N)
  const int hh = lane >> 4;    // K sub-group select
  const int cb = wave * 16;
  v8f acc = {};
  for (int kc = 0; kc < FDIM; kc += 64) {
    __syncthreads();
    stage_w64(wl, W1, kc, 64, t);
    __syncthreads();
#pragma unroll
    for (int k = 0; k < 64; k += 4) {
      const int kk = k + 2 * hh;
      v2f a, b;
      a.x = xs[n * LSTR + kc + kk];
      a.y = xs[n * LSTR + kc + kk + 1];
      b.x = wl[kk * WSTR + cb + n];
      b.y = wl[(kk + 1) * WSTR + cb + n];
      acc = wmma4(a, b, acc);
    }
  }
  const float bias = b1[cb + n];
#pragma unroll
  for (int r = 0; r < 8; ++r)
    h1[(row0 + r + 8 * hh) * FDIM + cb + n] = acc[r] + bias;
}

// -------------------------------------- column sums / sumsq over N (atomics)
__global__ __launch_bounds__(256) void k_colstats(const float* __restrict__ src,
                                                  int nrows,
                                                  float* __restrict__ sum,
                                                  float* __restrict__ sumsq) {
  const int t = threadIdx.x;
  const int c = t & 127;
  const int r0 = blockIdx.x * 2 + (t >> 7);
  const int stride = gridDim.x * 2;
  float s = 0.f, q = 0.f;
  for (int r = r0; r < nrows; r += stride) {
    float v = src[r * FDIM + c];
    s += v;
    q += v * v;
  }
  atomicAdd(&sum[c], s);
  atomicAdd(&sumsq[c], q);
}

__global__ void k_finstats(const float* __restrict__ sum,
                           const float* __restrict__ sumsq, float inv_n,
                           float* __restrict__ mean, float* __restrict__ rstd) {
  const int j = threadIdx.x;  // 128 threads
  float m = sum[j] * inv_n;
  float v = sumsq[j] * inv_n - m * m;
  v = v > 0.f ? v : 0.f;
  mean[j] = m;
  rstd[j] = 1.f / (sqrtf(v) + 1e-5f);
}

// --- GEMM2: sfeat = elu(ln(h1)) @ W2 + b2 + src_data @ Ws + bs  (fused dual-K)
__global__ __launch_bounds__(256) void k_gemm_sfeat(
    const float* __restrict__ h1, const float* __restrict__ mean,
    const float* __restrict__ rstd, const float* __restrict__ g,
    const float* __restrict__ be, const float* __restrict__ W2,
    const float* __restrict__ b2, const float* __restrict__ src_data,
    const float* __restrict__ Ws, const float* __restrict__ bs,
    float* __restrict__ sfeat) {
  __shared__ float hs[16 * LSTR];
  __shared__ float ss[16 * 36];
  __shared__ float wl[64 * WSTR];
  const int t = threadIdx.x;
  const int row0 = blockIdx.x * 16;
#pragma unroll
  for (int i = 0; i < 8; ++i) {
    int idx = t + i * 256;
    int r = idx >> 7, c = idx & 127;
    float v = h1[(row0 + r) * FDIM + c];
    v = (v - mean[c]) * rstd[c] * g[c] + be[c];
    hs[r * LSTR + c] = elu1(v);
  }
#pragma unroll
  for (int i = 0; i < 2; ++i) {
    int idx = t + i * 256;  // 512 = 16 rows x 32 cols
    int r = idx >> 5, c = idx & 31;
    ss[r * 36 + c] = src_data[(row0 + r) * 32 + c];
  }
  const int wave = t >> 5;
  const int lane = t & 31;
  const int n = lane & 15;
  const int hh = lane >> 4;
  const int cb = wave * 16;
  v8f acc = {};
  for (int kc = 0; kc < FDIM; kc += 64) {
    __syncthreads();
    stage_w64(wl, W2, kc, 64, t);
    __syncthreads();
#pragma unroll
    for (int k = 0; k < 64; k += 4) {
      const int kk = k + 2 * hh;
      v2f a, b;
      a.x = hs[n * LSTR + kc + kk];
      a.y = hs[n * LSTR + kc + kk + 1];
      b.x = wl[kk * WSTR + cb + n];
      b.y = wl[(kk + 1) * WSTR + cb + n];
      acc = wmma4(a, b, acc);
    }
  }
  __syncthreads();
  stage_w64(wl, Ws, 0, 32, t);  // lin_src_W is 32x128
  __syncthreads();
#pragma unroll
  for (int k = 0; k < 32; k += 4) {
    const int kk = k + 2 * hh;
    v2f a, b;
    a.x = ss[n * 36 + kk];
    a.y = ss[n * 36 + kk + 1];
    b.x = wl[kk * WSTR + cb + n];
    b.y = wl[(kk + 1) * WSTR + cb + n];
    acc = wmma4(a, b, acc);
  }
  const float bias = b2[cb + n] + bs[cb + n];
#pragma unroll
  for (int r = 0; r < 8; ++r)
    sfeat[(row0 + r + 8 * hh) * FDIM + cb + n] = acc[r] + bias;
}

// -------------------- edge Gram: s = sum(edata), G = edata^T edata (upper tri)
__global__ __launch_bounds__(256) void k_edge_gram(const float* __restrict__ edata,
                                                   int ne, float* __restrict__ s4,
                                                   float* __restrict__ G) {
  const int tid = blockIdx.x * 256 + threadIdx.x;
  const int stride = gridDim.x * 256;
  float s0 = 0, s1 = 0, s2 = 0, s3 = 0;
  float g[10];
#pragma unroll
  for (int i = 0; i < 10; ++i) g[i] = 0.f;
  for (int e = tid; e < ne; e += stride) {
    float4 d = ((const float4*)edata)[e];
    s0 += d.x; s1 += d.y; s2 += d.z; s3 += d.w;
    g[0] += d.x * d.x; g[1] += d.x * d.y; g[2] += d.x * d.z; g[3] += d.x * d.w;
    g[4] += d.y * d.y; g[5] += d.y * d.z; g[6] += d.y * d.w;
    g[7] += d.z * d.z; g[8] += d.z * d.w; g[9] += d.w * d.w;
  }
#pragma unroll
  for (int off = 16; off > 0; off >>= 1) {
    s0 += __shfl_xor(s0, off); s1 += __shfl_xor(s1, off);
    s2 += __shfl_xor(s2, off); s3 += __shfl_xor(s3, off);
#pragma unroll
    for (int i = 0; i < 10; ++i) g[i] += __shfl_xor(g[i], off);
  }
  if ((threadIdx.x & 31) == 0) {
    atomicAdd(&s4[0], s0); atomicAdd(&s4[1], s1);
    atomicAdd(&s4[2], s2); atomicAdd(&s4[3], s3);
#pragma unroll
    for (int i = 0; i < 10; ++i) atomicAdd(&G[i], g[i]);
  }
}

// analytic column-LN stats of (edata @ eW1 + eb1) over E, from s and Gram
__global__ void k_edge_stats(const float* __restrict__ s4,
                             const float* __restrict__ G,
                             const float* __restrict__ eW1,
                             const float* __restrict__ eb1, float inv_e,
                             float* __restrict__ emu, float* __restrict__ erstd) {
  const int j = threadIdx.x;  // 128
  float w0 = eW1[0 * FDIM + j], w1 = eW1[1 * FDIM + j];
  float w2 = eW1[2 * FDIM + j], w3 = eW1[3 * FDIM + j];
  float b = eb1[j];
  float sw = s4[0] * w0 + s4[1] * w1 + s4[2] * w2 + s4[3] * w3;
  float q = G[0] * w0 * w0 + G[4] * w1 * w1 + G[7] * w2 * w2 + G[9] * w3 * w3 +
            2.f * (G[1] * w0 * w1 + G[2] * w0 * w2 + G[3] * w0 * w3 +
                   G[5] * w1 * w2 + G[6] * w1 * w3 + G[8] * w2 * w3);
  float m = sw * inv_e + b;
  float eh2 = (q + 2.f * b * sw) * inv_e + b * b;
  float v = eh2 - m * m;
  v = v > 0.f ? v : 0.f;
  emu[j] = m;
  erstd[j] = 1.f / (sqrtf(v) + 1e-5f);
}

// per-edge logit + ordered-uint segment max (LN affine folded into weights)
__global__ __launch_bounds__(256) void k_logits(
    const float* __restrict__ edata, const int* __restrict__ dst_idx,
    const float* __restrict__ eW1, const float* __restrict__ eb1,
    const float* __restrict__ eg, const float* __restrict__ ebe,
    const float* __restrict__ eW2, const float* __restrict__ eb2,
    const float* __restrict__ emu, const float* __restrict__ erstd,
    float* __restrict__ logits, unsigned int* __restrict__ segmax) {
  __shared__ float w1s[4 * FDIM], w2s[FDIM], bcs[FDIM];
  const int t = threadIdx.x;
  if (t < FDIM) {
    const float A = erstd[t] * eg[t];  // rstd*g
    w1s[0 * FDIM + t] = eW1[0 * FDIM + t] * A;
    w1s[1 * FDIM + t] = eW1[1 * FDIM + t] * A;
    w1s[2 * FDIM + t] = eW1[2 * FDIM + t] * A;
    w1s[3 * FDIM + t] = eW1[3 * FDIM + t] * A;
    bcs[t] = (eb1[t] - emu[t]) * A + ebe[t];
    w2s[t] = eW2[t];
  }
  __syncthreads();
  const int e = blockIdx.x * 256 + t;
  float4 d = ((const float4*)edata)[e];
  float acc = eb2[0];
#pragma unroll 4
  for (int j = 0; j < FDIM; ++j) {
    float v = d.x * w1s[j] + d.y * w1s[FDIM + j] + d.z * w1s[2 * FDIM + j] +
              d.w * w1s[3 * FDIM + j] + bcs[j];
    acc += elu1(v) * w2s[j];
  }
  logits[e] = acc;
  unsigned ub = __float_as_uint(acc);
  unsigned u = (ub & 0x80000000u) ? ~ub : (ub | 0x80000000u);
  atomicMax(&segmax[dst_idx[e]], u);
}

// ex = exp(logit - max[dst]) (in place), den[dst] += ex
__global__ __launch_bounds__(256) void k_exp(const int* __restrict__ dst_idx,
                                             const unsigned int* __restrict__ segmax,
                                             float* __restrict__ logits,
                                             float* __restrict__ den) {
  const int e = blockIdx.x * 256 + threadIdx.x;
  const int d = dst_idx[e];
  unsigned u = segmax[d];
  unsigned b = (u & 0x80000000u) ? (u & 0x7FFFFFFFu) : ~u;
  float mx = __uint_as_float(b);
  float ex = expf(logits[e] - mx);
  logits[e] = ex;
  atomicAdd(&den[d], ex);
}

// nf[dst] += (ex/den[dst]) * sfeat[src]   (2 edges per 256-thread block)
__global__ __launch_bounds__(256) void k_scatter(const int* __restrict__ src_idx,
                                                 const int* __restrict__ dst_idx,
                                                 const float* __restrict__ ex,
                                                 const float* __restrict__ den,
                                                 const float* __restrict__ sfeat,
                                                 float* __restrict__ nf) {
  const int t = threadIdx.x;
  const int e = blockIdx.x * 2 + (t >> 7);
  const int f = t & 127;
  const int d = dst_idx[e], s = src_idx[e];
  const float w = ex[e] / den[d];
  atomicAdd(&nf[d * FDIM + f], sfeat[s * FDIM + f] * w);
}

// final: out = elu(ln(nf)) @ out_W + out_b  (in place on d_out, row-local)
__global__ __launch_bounds__(256) void k_out(const float* __restrict__ mean,
                                             const float* __restrict__ rstd,
                                             const float* __restrict__ g,
                                             const float* __restrict__ be,
                                             const float* __restrict__ W,
                                             const float* __restrict__ bo,
                                             float* __restrict__ out) {
  __shared__ float ns[16 * LSTR];
  __shared__ float wl[64 * WSTR];
  const int t = threadIdx.x;
  const int row0 = blockIdx.x * 16;
#pragma unroll
  for (int i = 0; i < 8; ++i) {
    int idx = t + i * 256;
    int r = idx >> 7, c = idx & 127;
    float v = out[(row0 + r) * FDIM + c];
    v = (v - mean[c]) * rstd[c] * g[c] + be[c];
    ns[r * LSTR + c] = elu1(v);
  }
  const int wave = t >> 5;
  const int lane = t & 31;
  const int n = lane & 15;
  const int hh = lane >> 4;
  const int cb = wave * 16;
  v8f acc = {};
  for (int kc = 0; kc < FDIM; kc += 64) {
    __syncthreads();
    stage_w64(wl, W, kc, 64, t);
    __syncthreads();
#pragma unroll
    for (int k = 0; k < 64; k += 4) {
      const int kk = k + 2 * hh;
      v2f a, b;
      a.x = ns[n * LSTR + kc + kk];
      a.y = ns[n * LSTR + kc + kk + 1];
      b.x = wl[kk * WSTR + cb + n];
      b.y = wl[(kk + 1) * WSTR + cb + n];
      acc = wmma4(a, b, acc);
    }
  }
  const float bias = bo[cb + n];
#pragma unroll
  for (int r = 0; r < 8; ++r)
    out[(row0 + r + 8 * hh) * FDIM + cb + n] = acc[r] + bias;
}

// ---------------------------------------------------------------------------
extern "C" void kernel_launch(void* const* d_in, const int* in_sizes, int n_in,
                              void* d_out, int out_size, void* d_ws,
                              size_t ws_size, hipStream_t stream) {
  const float* x         = (const float*)d_in[0];
  const float* src_data  = (const float*)d_in[1];
  const float* edata     = (const float*)d_in[2];
  const int*   src_idx   = (const int*)d_in[3];
  const int*   dst_idx   = (const int*)d_in[4];
  const float* lin_src_W = (const float*)d_in[5];
  const float* lin_src_b = (const float*)d_in[6];
  const float* node_W1   = (const float*)d_in[7];
  const float* node_b1   = (const float*)d_in[8];
  const float* node_g    = (const float*)d_in[9];
  const float* node_be   = (const float*)d_in[10];
  const float* node_W2   = (const float*)d_in[11];
  const float* node_b2   = (const float*)d_in[12];
  const float* edge_W1   = (const float*)d_in[13];
  const float* edge_b1   = (const float*)d_in[14];
  const float* edge_g    = (const float*)d_in[15];
  const float* edge_be   = (const float*)d_in[16];
  const float* edge_W2   = (const float*)d_in[17];
  const float* edge_b2   = (const float*)d_in[18];
  const float* out_g     = (const float*)d_in[19];
  const float* out_be    = (const float*)d_in[20];
  const float* out_W     = (const float*)d_in[21];
  const float* out_b     = (const float*)d_in[22];
  float* out = (float*)d_out;

  float* ws = (float*)d_ws;
  float* h1     = ws;                      // N*128
  float* sfeat  = ws + 6400000;            // N*128
  float* logits = ws + 12800000;           // E
  unsigned* segmax = (unsigned*)(ws + 13600000);  // N
  float* den    = ws + 13650000;           // N
  float* sum    = ws + 13700000;           // 128
  float* sumsq  = sum + 128;               // 128
  float* mean   = sumsq + 128;             // 128
  float* rstd   = mean + 128;              // 128
  float* es     = rstd + 128;              // 4
  float* eG     = es + 4;                  // 16 (10 used)
  float* emu    = eG + 16;                 // 128
  float* erstd  = emu + 128;               // 128

  // deterministic per-call init of all accumulators
  hipMemsetAsync(out, 0, (size_t)N_NODES * FDIM * sizeof(float), stream);
  hipMemsetAsync(segmax, 0, (size_t)N_NODES * sizeof(unsigned), stream);
  hipMemsetAsync(den, 0, (size_t)N_NODES * sizeof(float), stream);
  hipMemsetAsync(sum, 0, 256 * sizeof(float), stream);
  hipMemsetAsync(es, 0, 20 * sizeof(float), stream);

  k_gemm_h1<<<N_NODES / 16, 256, 0, stream>>>(x, node_W1, node_b1, h1);
  k_colstats<<<200, 256, 0, stream>>>(h1, N_NODES, sum, sumsq);
  k_finstats<<<1, 128, 0, stream>>>(sum, sumsq, 1.f / N_NODES, mean, rstd);
  k_gemm_sfeat<<<N_NODES / 16, 256, 0, stream>>>(h1, mean, rstd, node_g, node_be,
                                                 node_W2, node_b2, src_data,
                                                 lin_src_W, lin_src_b, sfeat);
  k_edge_gram<<<400, 256, 0, stream>>>(edata, N_EDGES, es, eG);
  k_edge_stats<<<1, 128, 0, stream>>>(es, eG, edge_W1, edge_b1, 1.f / N_EDGES,
                                      emu, erstd);
  k_logits<<<N_EDGES / 256, 256, 0, stream>>>(edata, dst_idx, edge_W1, edge_b1,
                                              edge_g, edge_be, edge_W2, edge_b2,
                                              emu, erstd, logits, segmax);
  k_exp<<<N_EDGES / 256, 256, 0, stream>>>(dst_idx, segmax, logits, den);
  k_scatter<<<N_EDGES / 2, 256, 0, stream>>>(src_idx, dst_idx, logits, den,
                                             sfeat, out);
  hipMemsetAsync(sum, 0, 256 * sizeof(float), stream);
  k_colstats<<<200, 256, 0, stream>>>(out, N_NODES, sum, sumsq);
  k_finstats<<<1, 128, 0, stream>>>(sum, sumsq, 1.f / N_NODES, mean, rstd);
  k_out<<<N_NODES / 16, 256, 0, stream>>>(mean, rstd, out_g, out_be, out_W,
                                          out_b, out);
}